// VolatileMemoryController_32091995636215
// MI455X (gfx1250) — compile-verified
//
#include <hip/hip_runtime.h>
#include <math.h>

typedef float v2f __attribute__((ext_vector_type(2)));
typedef float v8f __attribute__((ext_vector_type(8)));

#define D_MODEL 1024
#define D_SLOT  128
#define KSLOT   8
#define HDIM    512
#define BATCH   4
#define SEQ     4096
#define NTOK    (BATCH*SEQ)
#define SCALEQ  0.08838834764831845f   /* 1/sqrt(128) */

// ---------------------------------------------------------------------------
// f32 WMMA: D = A(16x4) x B(4x16) + C(16x16)
//  A: lane m=lane%16; vgpr0/1 = K (half*2 +0/+1)
//  B: vgpr v = row K=(half*2+v), col n=lane%16
//  C: vgpr v = row (v+half*8), col n=lane%16
// ---------------------------------------------------------------------------
__device__ __forceinline__ v8f wmma_acc(v2f a, v2f b, v8f acc) {
  return __builtin_amdgcn_wmma_f32_16x16x4_f32(false, a, false, b,
                                               (short)0, acc, false, false);
}

__device__ __forceinline__ float gelu_exact(float x) {
  return 0.5f * x * (1.0f + erff(x * 0.70710678118654752440f));
}
__device__ __forceinline__ float sigmoidf_(float x) {
  return 1.0f / (1.0f + expf(-x));
}

// ---------------------------------------------------------------------------
// Blocked + software-pipelined f32 WMMA GEMM.
// One wave computes a 64x32 block (4 M-tiles x 2 N-tiles); fragments for
// iteration kk+4 are prefetched into separate registers before the 8 WMMAs
// of iteration kk issue, so load latency retires behind the WMMA burst.
// ---------------------------------------------------------------------------
__global__ void gemm_wmma_kernel(const float* __restrict__ X, int ldx,
                                 const float* __restrict__ W, int ldw,
                                 const float* __restrict__ bias,
                                 float* __restrict__ Y, int ldy,
                                 int Kdim, int tiles_m, int tiles_n) {
  const int mg = tiles_m >> 2, ng = tiles_n >> 1;
  int wave = blockIdx.x * (blockDim.x >> 5) + (threadIdx.x >> 5);
  if (wave >= mg * ng) return;                 // wave-uniform
  int tm4 = wave % mg, tn2 = wave / mg;
  int lane = threadIdx.x & 31, half = lane >> 4, l16 = lane & 15;
  int rbase = tm4 * 64, cbase = tn2 * 32;

  const float* xr[4];
#pragma unroll
  for (int i = 0; i < 4; ++i)
    xr[i] = X + (size_t)(rbase + i * 16 + l16) * ldx + half * 2;
  const float* wc0 = W + (size_t)(half * 2) * ldw + cbase + l16;

  v8f acc[4][2] = {};
  v2f a[4], b[2];
#pragma unroll
  for (int i = 0; i < 4; ++i) a[i] = *(const v2f*)(xr[i]);
  b[0].x = wc0[0];  b[0].y = wc0[ldw];
  b[1].x = wc0[16]; b[1].y = wc0[16 + ldw];

#pragma unroll 2
  for (int kk = 0; kk < Kdim - 4; kk += 4) {
    int kn = kk + 4;
    v2f an[4], bn[2];
#pragma unroll
    for (int i = 0; i < 4; ++i) an[i] = *(const v2f*)(xr[i] + kn);
    const float* wk = wc0 + (size_t)kn * ldw;
    bn[0].x = wk[0];  bn[0].y = wk[ldw];
    bn[1].x = wk[16]; bn[1].y = wk[16 + ldw];
#pragma unroll
    for (int i = 0; i < 4; ++i)
#pragma unroll
      for (int j = 0; j < 2; ++j)
        acc[i][j] = wmma_acc(a[i], b[j], acc[i][j]);
#pragma unroll
    for (int i = 0; i < 4; ++i) a[i] = an[i];
    b[0] = bn[0]; b[1] = bn[1];
  }
  // drain
#pragma unroll
  for (int i = 0; i < 4; ++i)
#pragma unroll
    for (int j = 0; j < 2; ++j)
      acc[i][j] = wmma_acc(a[i], b[j], acc[i][j]);

#pragma unroll
  for (int j = 0; j < 2; ++j) {
    int col = cbase + j * 16 + l16;
    float bv = bias ? bias[col] : 0.0f;
#pragma unroll
    for (int i = 0; i < 4; ++i)
#pragma unroll
      for (int v = 0; v < 8; ++v) {
        int r = rbase + i * 16 + v + half * 8;
        Y[(size_t)r * ldy + col] = acc[i][j][v] + bv;
      }
  }
}

// ---------------------------------------------------------------------------
// Fused 2-layer MLP, 4 M-tiles per wave, pipelined K loop:
// out[row] = f( w2 . gelu(Xcat @ W1 + b1) + b2 ),  Xcat=[X1 (K1) | X2 (K2)]
// mode: 0=raw 1=sigmoid 2=sigmoid(/temp)
// ---------------------------------------------------------------------------
__global__ void mlp_wmma_kernel(const float* __restrict__ X1, int K1,
                                const float* __restrict__ X2, int K2, int ld2,
                                const float* __restrict__ W1, int ldw,
                                const float* __restrict__ b1,
                                const float* __restrict__ w2,
                                const float* __restrict__ b2,
                                const float* __restrict__ temp_ptr,
                                int mode, float* __restrict__ out,
                                int tiles_m) {
  const int mg = tiles_m >> 2;
  int wave = blockIdx.x * (blockDim.x >> 5) + (threadIdx.x >> 5);
  if (wave >= mg) return;
  int lane = threadIdx.x & 31, half = lane >> 4, l16 = lane & 15;
  int rbase = wave * 64;
  int Ktot = K1 + K2;
  int ntiles = ldw >> 4;                       // H/16

  const float* x1r[4];
  const float* x2r[4];
#pragma unroll
  for (int i = 0; i < 4; ++i) {
    int row = rbase + i * 16 + l16;
    x1r[i] = X1 + (size_t)row * K1 + half * 2;
    x2r[i] = X2 ? (X2 + (size_t)row * ld2 + half * 2) : x1r[i];
  }

  float partial[4][8] = {};
  for (int nt = 0; nt < ntiles; ++nt) {
    int ncol = nt * 16 + l16;
    v8f acc[4] = {};
    v2f a[4], b;
#pragma unroll
    for (int i = 0; i < 4; ++i) a[i] = *(const v2f*)(x1r[i]);
    {
      const float* wc = W1 + (size_t)(half * 2) * ldw + ncol;
      b.x = wc[0]; b.y = wc[ldw];
    }
#pragma unroll 2
    for (int kk = 0; kk < Ktot - 4; kk += 4) {
      int kn = kk + 4;
      v2f an[4], bn;
      const float* wc = W1 + (size_t)(kn + half * 2) * ldw + ncol;
      bn.x = wc[0]; bn.y = wc[ldw];
#pragma unroll
      for (int i = 0; i < 4; ++i)
        an[i] = (kn < K1) ? *(const v2f*)(x1r[i] + kn)
                          : *(const v2f*)(x2r[i] + (kn - K1));
#pragma unroll
      for (int i = 0; i < 4; ++i) acc[i] = wmma_acc(a[i], b, acc[i]);
#pragma unroll
      for (int i = 0; i < 4; ++i) a[i] = an[i];
      b = bn;
    }
#pragma unroll
    for (int i = 0; i < 4; ++i) acc[i] = wmma_acc(a[i], b, acc[i]);

    float w2v = w2[ncol];
    float b1v = b1[ncol];
#pragma unroll
    for (int i = 0; i < 4; ++i)
#pragma unroll
      for (int v = 0; v < 8; ++v)
        partial[i][v] += gelu_exact(acc[i][v] + b1v) * w2v;
  }
  // reduce over 16 lanes within each half-wave (bit4 = half, untouched)
#pragma unroll
  for (int i = 0; i < 4; ++i)
#pragma unroll
    for (int v = 0; v < 8; ++v) {
      float s = partial[i][v];
      s += __shfl_xor(s, 1, 32);
      s += __shfl_xor(s, 2, 32);
      s += __shfl_xor(s, 4, 32);
      s += __shfl_xor(s, 8, 32);
      partial[i][v] = s;
    }
  if (l16 == 0) {
    float b2v = b2[0];
    float temp = temp_ptr ? fmaxf(temp_ptr[0], 0.1f) : 1.0f;
#pragma unroll
    for (int i = 0; i < 4; ++i)
#pragma unroll
      for (int v = 0; v < 8; ++v) {
        float logit = partial[i][v] + b2v;
        float val = (mode == 0) ? logit
                  : (mode == 1) ? sigmoidf_(logit)
                                : sigmoidf_(logit / temp);
        out[rbase + i * 16 + v + half * 8] = val;
      }
  }
}

// ---------------------------------------------------------------------------
// Fusion + residual gate (64x32 block per wave, pipelined):
//  fused = x @ Wf[0:1024] + ctx128 @ G + beff ;  out = (1-g)*x + g*fused
// ---------------------------------------------------------------------------
__global__ void fusion_wmma_kernel(const float* __restrict__ X,
                                   const float* __restrict__ CTX,
                                   const float* __restrict__ Wf,
                                   const float* __restrict__ G,
                                   const float* __restrict__ beff,
                                   const float* __restrict__ gate,
                                   float* __restrict__ out, int tiles_m) {
  const int tiles_n = D_MODEL / 16;
  const int mg = tiles_m >> 2, ng = tiles_n >> 1;
  int wave = blockIdx.x * (blockDim.x >> 5) + (threadIdx.x >> 5);
  if (wave >= mg * ng) return;
  int tm4 = wave % mg, tn2 = wave / mg;
  int lane = threadIdx.x & 31, half = lane >> 4, l16 = lane & 15;
  int rbase = tm4 * 64, cbase = tn2 * 32;

  const float* xr[4];
  const float* cr[4];
#pragma unroll
  for (int i = 0; i < 4; ++i) {
    int row = rbase + i * 16 + l16;
    xr[i] = X + (size_t)row * D_MODEL + half * 2;
    cr[i] = CTX + (size_t)row * D_SLOT + half * 2;
  }
  const float* wc0 = Wf + (size_t)(half * 2) * D_MODEL + cbase + l16;
  const float* gc0 = G + (size_t)(half * 2) * D_MODEL + cbase + l16;

  v8f acc[4][2] = {};
  v2f a[4], b[2];

  // ---- segment 1: x @ Wf  (K = 1024) ----
#pragma unroll
  for (int i = 0; i < 4; ++i) a[i] = *(const v2f*)(xr[i]);
  b[0].x = wc0[0];  b[0].y = wc0[D_MODEL];
  b[1].x = wc0[16]; b[1].y = wc0[16 + D_MODEL];
#pragma unroll 2
  for (int kk = 0; kk < D_MODEL - 4; kk += 4) {
    int kn = kk + 4;
    v2f an[4], bn[2];
#pragma unroll
    for (int i = 0; i < 4; ++i) an[i] = *(const v2f*)(xr[i] + kn);
    const float* wk = wc0 + (size_t)kn * D_MODEL;
    bn[0].x = wk[0];  bn[0].y = wk[D_MODEL];
    bn[1].x = wk[16]; bn[1].y = wk[16 + D_MODEL];
#pragma unroll
    for (int i = 0; i < 4; ++i)
#pragma unroll
      for (int j = 0; j < 2; ++j)
        acc[i][j] = wmma_acc(a[i], b[j], acc[i][j]);
#pragma unroll
    for (int i = 0; i < 4; ++i) a[i] = an[i];
    b[0] = bn[0]; b[1] = bn[1];
  }
#pragma unroll
  for (int i = 0; i < 4; ++i)
#pragma unroll
    for (int j = 0; j < 2; ++j)
      acc[i][j] = wmma_acc(a[i], b[j], acc[i][j]);

  // ---- segment 2: ctx128 @ G  (K = 128) ----
#pragma unroll
  for (int i = 0; i < 4; ++i) a[i] = *(const v2f*)(cr[i]);
  b[0].x = gc0[0];  b[0].y = gc0[D_MODEL];
  b[1].x = gc0[16]; b[1].y = gc0[16 + D_MODEL];
#pragma unroll 2
  for (int kk = 0; kk < D_SLOT - 4; kk += 4) {
    int kn = kk + 4;
    v2f an[4], bn[2];
#pragma unroll
    for (int i = 0; i < 4; ++i) an[i] = *(const v2f*)(cr[i] + kn);
    const float* gk = gc0 + (size_t)kn * D_MODEL;
    bn[0].x = gk[0];  bn[0].y = gk[D_MODEL];
    bn[1].x = gk[16]; bn[1].y = gk[16 + D_MODEL];
#pragma unroll
    for (int i = 0; i < 4; ++i)
#pragma unroll
      for (int j = 0; j < 2; ++j)
        acc[i][j] = wmma_acc(a[i], b[j], acc[i][j]);
#pragma unroll
    for (int i = 0; i < 4; ++i) a[i] = an[i];
    b[0] = bn[0]; b[1] = bn[1];
  }
#pragma unroll
  for (int i = 0; i < 4; ++i)
#pragma unroll
    for (int j = 0; j < 2; ++j)
      acc[i][j] = wmma_acc(a[i], b[j], acc[i][j]);

#pragma unroll
  for (int j = 0; j < 2; ++j) {
    int col = cbase + j * 16 + l16;
    float bv = beff[col];
#pragma unroll
    for (int i = 0; i < 4; ++i)
#pragma unroll
      for (int v = 0; v < 8; ++v) {
        int r = rbase + i * 16 + v + half * 8;
        float g = gate[r];
        float xv = X[(size_t)r * D_MODEL + col];
        out[(size_t)r * D_MODEL + col] = (1.0f - g) * xv + g * (acc[i][j][v] + bv);
      }
  }
}

// ---------------------------------------------------------------------------
// Per-token 8-slot attention (read + write query paths)
// ---------------------------------------------------------------------------
__global__ void attn_kernel(const float* __restrict__ wm,
                            const float* __restrict__ q,
                            const float* __restrict__ wq,
                            float* __restrict__ attn_out,
                            float* __restrict__ ctx128,
                            float* __restrict__ wscores_out,
                            float* __restrict__ wmctx) {
  int t = blockIdx.x * blockDim.x + threadIdx.x;
  if (t >= NTOK) return;
  int b = t / SEQ;
  const float* wmb = wm + (size_t)b * KSLOT * (D_SLOT + 1);
  const float* qp  = q  + (size_t)t * D_SLOT;
  const float* wqp = wq + (size_t)t * D_SLOT;

  float rs[KSLOT], wsc[KSLOT];
#pragma unroll
  for (int j = 0; j < KSLOT; ++j) {
    const float* c = wmb + j * (D_SLOT + 1);
    float d1 = 0.f, d2 = 0.f;
#pragma unroll 4
    for (int d = 0; d < D_SLOT; ++d) { float cv = c[d]; d1 += qp[d] * cv; d2 += wqp[d] * cv; }
    float s1 = d1 * SCALEQ;
    if (c[D_SLOT] < 0.1f) s1 = -1.0e9f;       // stale mask (read path only)
    rs[j] = s1;
    wsc[j] = d2 * SCALEQ;
    wscores_out[(size_t)t * KSLOT + j] = wsc[j];
  }
  float m1 = rs[0], m2 = wsc[0];
#pragma unroll
  for (int j = 1; j < KSLOT; ++j) { m1 = fmaxf(m1, rs[j]); m2 = fmaxf(m2, wsc[j]); }
  float a1[KSLOT], a2[KSLOT], e1 = 0.f, e2 = 0.f;
#pragma unroll
  for (int j = 0; j < KSLOT; ++j) {
    a1[j] = expf(rs[j] - m1);  e1 += a1[j];
    a2[j] = expf(wsc[j] - m2); e2 += a2[j];
  }
#pragma unroll
  for (int j = 0; j < KSLOT; ++j) {
    a1[j] /= e1; a2[j] /= e2;
    attn_out[(size_t)t * KSLOT + j] = a1[j];
  }
#pragma unroll 4
  for (int d = 0; d < D_SLOT; ++d) {
    float s1 = 0.f, s2 = 0.f;
#pragma unroll
    for (int j = 0; j < KSLOT; ++j) {
      float cv = wmb[j * (D_SLOT + 1) + d];
      s1 += a1[j] * cv; s2 += a2[j] * cv;
    }
    ctx128[(size_t)t * D_SLOT + d] = s1;
    wmctx [(size_t)t * D_SLOT + d] = s2;
  }
}

// ---------------------------------------------------------------------------
// Per-batch reductions over S: read pressure -> new freshness; importance sum.
// ---------------------------------------------------------------------------
__global__ void reduce_kernel(const float* __restrict__ attn,
                              const float* __restrict__ imp_logit,
                              const float* __restrict__ wm,
                              const float* __restrict__ wtemp,
                              float* __restrict__ nf,
                              float* __restrict__ impsum) {
  int b = blockIdx.x, tid = threadIdx.x;
  float rp[KSLOT] = {0, 0, 0, 0, 0, 0, 0, 0};
  float es = 0.f;
  float temp = fmaxf(wtemp[0], 0.1f);
  for (int s = tid; s < SEQ; s += blockDim.x) {
    size_t t = (size_t)b * SEQ + s;
#pragma unroll
    for (int j = 0; j < KSLOT; ++j) rp[j] += attn[t * KSLOT + j];
    es += expf(imp_logit[t] / temp);
  }
  __shared__ float red[256];
  __shared__ float res[KSLOT + 1];
  for (int j = 0; j <= KSLOT; ++j) {
    red[tid] = (j < KSLOT) ? rp[j] : es;
    __syncthreads();
    for (int off = 128; off > 0; off >>= 1) {
      if (tid < off) red[tid] += red[tid + off];
      __syncthreads();
    }
    if (tid == 0) res[j] = red[0];
    __syncthreads();
  }
  if (tid == 0) {
    float maxp = 1e-8f;
    for (int j = 0; j < KSLOT; ++j) maxp = fmaxf(maxp, res[j]);
    for (int j = 0; j < KSLOT; ++j) {
      float fr = wm[(size_t)b * KSLOT * (D_SLOT + 1) + j * (D_SLOT + 1) + D_SLOT];
      nf[b * KSLOT + j] = fr * (1.0f - (res[j] / maxp) * 0.7f);
    }
    impsum[b] = res[KSLOT];
  }
}

// ---------------------------------------------------------------------------
// Per-token write path: slot argmax, one-hot, accumulate new content.
// ---------------------------------------------------------------------------
__global__ void write_kernel(const float* __restrict__ wscores,
                             const float* __restrict__ nf,
                             const float* __restrict__ imp_logit,
                             const float* __restrict__ impsum,
                             const float* __restrict__ wtemp,
                             const float* __restrict__ wgate,
                             const float* __restrict__ xslot,
                             float* __restrict__ slotsel_out,
                             float* __restrict__ acc_content,
                             float* __restrict__ slot_total) {
  int t = blockIdx.x * blockDim.x + threadIdx.x;
  if (t >= NTOK) return;
  int b = t / SEQ;
  float temp = fmaxf(wtemp[0], 0.1f);
  float impv = expf(imp_logit[t] / temp) / (impsum[b] + 1e-8f) * (float)SEQ;
  int best = 0;
  float bv = -3.4e38f;
#pragma unroll
  for (int j = 0; j < KSLOT; ++j) {
    float sc = wscores[(size_t)t * KSLOT + j] - 2.0f * nf[b * KSLOT + j];
    slotsel_out[(size_t)t * KSLOT + j] = 0.0f;
    if (sc > bv) { bv = sc; best = j; }        // first max (jnp.argmax)
  }
  slotsel_out[(size_t)t * KSLOT + best] = 1.0f;
  float ww = wgate[t] * impv;
  const float* xs = xslot + (size_t)t * D_SLOT;
  float* ap = acc_content + (b * KSLOT + best) * D_SLOT;
#pragma unroll 4
  for (int d = 0; d < D_SLOT; ++d) atomicAdd(&ap[d], ww * xs[d]);
  atomicAdd(&slot_total[b * KSLOT + best], ww);
}

// ---------------------------------------------------------------------------
// Finalize wm_updated (B,K,129)
// ---------------------------------------------------------------------------
__global__ void finalize_kernel(const float* __restrict__ wm,
                                const float* __restrict__ nf,
                                const float* __restrict__ acc,
                                const float* __restrict__ st,
                                float* __restrict__ out_wm) {
  int i = blockIdx.x * blockDim.x + threadIdx.x;
  if (i >= BATCH * KSLOT * (D_SLOT + 1)) return;
  int bk = i / (D_SLOT + 1), d = i % (D_SLOT + 1);
  float total = st[bk];
  float has = (total > 0.1f) ? 1.0f : 0.0f;
  float val;
  if (d < D_SLOT) {
    float nc = acc[bk * D_SLOT + d] / fmaxf(total, 1e-8f);
    val = has * nc + (1.0f - has) * wm[(size_t)bk * (D_SLOT + 1) + d];
  } else {
    val = has + (1.0f - has) * nf[bk];
  }
  out_wm[i] = val;
}

// ---------------------------------------------------------------------------
// beff[c] = fusion_b[c] + sum_k from_wm_b[k] * fusion_w[1024+k][c]
// ---------------------------------------------------------------------------
__global__ void beff_kernel(const float* __restrict__ from_wm_b,
                            const float* __restrict__ fusion_w,
                            const float* __restrict__ fusion_b,
                            float* __restrict__ beff) {
  int c = blockIdx.x * blockDim.x + threadIdx.x;
  if (c >= D_MODEL) return;
  float s = fusion_b[c];
  for (int k = 0; k < D_MODEL; ++k)
    s += from_wm_b[k] * fusion_w[(size_t)(D_MODEL + k) * D_MODEL + c];
  beff[c] = s;
}

// ---------------------------------------------------------------------------
extern "C" void kernel_launch(void* const* d_in, const int* in_sizes, int n_in,
                              void* d_out, int out_size, void* d_ws, size_t ws_size,
                              hipStream_t stream) {
  (void)in_sizes; (void)n_in; (void)out_size; (void)ws_size;
  const float* x        = (const float*)d_in[0];
  const float* wm       = (const float*)d_in[1];
  const float* rq_w     = (const float*)d_in[2];
  const float* rq_b     = (const float*)d_in[3];
  const float* fwm_w    = (const float*)d_in[4];
  const float* fwm_b    = (const float*)d_in[5];
  const float* rg1_w    = (const float*)d_in[6];
  const float* rg1_b    = (const float*)d_in[7];
  const float* rg2_w    = (const float*)d_in[8];
  const float* rg2_b    = (const float*)d_in[9];
  const float* fusion_w = (const float*)d_in[10];
  const float* fusion_b = (const float*)d_in[11];
  const float* to_wm_w  = (const float*)d_in[12];
  const float* to_wm_b  = (const float*)d_in[13];
  const float* wq_w     = (const float*)d_in[14];
  const float* wq_b     = (const float*)d_in[15];
  const float* imp1_w   = (const float*)d_in[16];
  const float* imp1_b   = (const float*)d_in[17];
  const float* imp2_w   = (const float*)d_in[18];
  const float* imp2_b   = (const float*)d_in[19];
  const float* wd1_w    = (const float*)d_in[20];
  const float* wd1_b    = (const float*)d_in[21];
  const float* wd2_w    = (const float*)d_in[22];
  const float* wd2_b    = (const float*)d_in[23];
  const float* wtemp    = (const float*)d_in[24];

  float* out = (float*)d_out;
  float* out_xenh  = out;                 // B*S*1024 = 16777216
  float* out_wm    = out + 16777216;      // B*K*129  = 4128
  float* out_gate  = out + 16781344;      // B*S      = 16384
  float* out_wgate = out + 16797728;      // B*S      = 16384
  float* out_attn  = out + 16814112;      // B*S*K    = 131072
  float* out_sel   = out + 16945184;      // B*S*K    = 131072

  float* ws = (float*)d_ws;
  size_t off = 0;
  float* ws_q     = ws + off; off += (size_t)NTOK * D_SLOT;
  float* ws_wq    = ws + off; off += (size_t)NTOK * D_SLOT;
  float* ws_xs    = ws + off; off += (size_t)NTOK * D_SLOT;
  float* ws_ctx   = ws + off; off += (size_t)NTOK * D_SLOT;
  float* ws_wmctx = ws + off; off += (size_t)NTOK * D_SLOT;
  float* ws_wsc   = ws + off; off += (size_t)NTOK * KSLOT;
  float* ws_imp   = ws + off; off += NTOK;
  float* ws_G     = ws + off; off += (size_t)D_SLOT * D_MODEL;
  float* ws_beff  = ws + off; off += D_MODEL;
  float* ws_nf    = ws + off; off += BATCH * KSLOT;
  float* ws_isum  = ws + off; off += BATCH;
  float* ws_accC  = ws + off; off += BATCH * KSLOT * D_SLOT;
  float* ws_stot  = ws + off; off += BATCH * KSLOT;

  hipMemsetAsync(ws_accC, 0, (BATCH * KSLOT * D_SLOT + BATCH * KSLOT) * sizeof(float), stream);

  dim3 blk(128);             // 4 waves/block
  const int TM = NTOK / 16;  // 1024 token tiles

  // 128-wide projections: q, wq, x_slot  -> (1024/4)*(8/2)=1024 waves
  gemm_wmma_kernel<<<dim3(256), blk, 0, stream>>>(x, D_MODEL, rq_w, D_SLOT, rq_b, ws_q, D_SLOT, D_MODEL, TM, 8);
  gemm_wmma_kernel<<<dim3(256), blk, 0, stream>>>(x, D_MODEL, wq_w, D_SLOT, wq_b, ws_wq, D_SLOT, D_MODEL, TM, 8);
  gemm_wmma_kernel<<<dim3(256), blk, 0, stream>>>(x, D_MODEL, to_wm_w, D_SLOT, to_wm_b, ws_xs, D_SLOT, D_MODEL, TM, 8);

  // G = from_wm_w @ fusion_w[1024:2048]  (128x1024) -> (8/4)*(64/2)=64 waves
  gemm_wmma_kernel<<<dim3(16), blk, 0, stream>>>(fwm_w, D_MODEL,
      fusion_w + (size_t)D_MODEL * D_MODEL, D_MODEL, (const float*)0, ws_G, D_MODEL, D_MODEL, 8, 64);
  beff_kernel<<<dim3(4), dim3(256), 0, stream>>>(fwm_b, fusion_w, fusion_b, ws_beff);

  // gate = sigmoid(MLP(x));  imp_logit = MLP(x)   -> 256 waves each
  mlp_wmma_kernel<<<dim3(64), blk, 0, stream>>>(x, D_MODEL, (const float*)0, 0, 0,
      rg1_w, HDIM, rg1_b, rg2_w, rg2_b, (const float*)0, 1, out_gate, TM);
  mlp_wmma_kernel<<<dim3(64), blk, 0, stream>>>(x, D_MODEL, (const float*)0, 0, 0,
      imp1_w, HDIM, imp1_b, imp2_w, imp2_b, (const float*)0, 0, ws_imp, TM);

  // slot attention (read softmax + ctx128, write scores + wm_ctx)
  attn_kernel<<<dim3(NTOK / 256), dim3(256), 0, stream>>>(wm, ws_q, ws_wq,
      out_attn, ws_ctx, ws_wsc, ws_wmctx);

  // per-batch reductions -> new freshness, importance normalizer
  reduce_kernel<<<dim3(BATCH), dim3(256), 0, stream>>>(out_attn, ws_imp, wm, wtemp, ws_nf, ws_isum);

  // write gate = sigmoid(MLP(concat(x, wm_ctx)) / temp)
  mlp_wmma_kernel<<<dim3(64), blk, 0, stream>>>(x, D_MODEL, ws_wmctx, D_SLOT, D_SLOT,
      wd1_w, HDIM, wd1_b, wd2_w, wd2_b, wtemp, 2, out_wgate, TM);

  // fused + gated residual -> x_enhanced  -> (1024/4)*(64/2)=8192 waves
  fusion_wmma_kernel<<<dim3(2048), blk, 0, stream>>>(x, ws_ctx, fusion_w,
      ws_G, ws_beff, out_gate, out_xenh, TM);

  // slot selection + content accumulation
  write_kernel<<<dim3(NTOK / 256), dim3(256), 0, stream>>>(ws_wsc, ws_nf, ws_imp,
      ws_isum, wtemp, out_wgate, ws_xs, out_sel, ws_accC, ws_stot);

  // wm_updated
  finalize_kernel<<<dim3((BATCH * KSLOT * (D_SLOT + 1) + 255) / 256), dim3(256), 0, stream>>>(
      wm, ws_nf, ws_accC, ws_stot, out_wm);
}